// GCN_4Layers_21388937134411
// MI455X (gfx1250) — compile-verified
//
#include <hip/hip_runtime.h>

#define N_NODES 50000
#define N_EDGES 800000
#define IN_F 512
#define H_F 128
#define OUT_F 64
#define NPAD 50048  // N_NODES padded to multiple of 64 for ws layout

typedef __attribute__((ext_vector_type(2))) float v2f;
typedef __attribute__((ext_vector_type(8))) float v8f;

// ---------------------------------------------------------------- utilities
__global__ void zero_kernel(float* __restrict__ p, int n) {
  int i = blockIdx.x * blockDim.x + threadIdx.x;
  if (i < n) p[i] = 0.0f;
}

__global__ void degree_kernel(const int* __restrict__ src, const int* __restrict__ dst,
                              float* __restrict__ odeg, float* __restrict__ ideg, int E) {
  int e = blockIdx.x * blockDim.x + threadIdx.x;
  if (e < E) {
    atomicAdd(&odeg[src[e]], 1.0f);
    atomicAdd(&ideg[dst[e]], 1.0f);
  }
}

__global__ void norm_kernel(float* __restrict__ p, int n) {
  int i = blockIdx.x * blockDim.x + threadIdx.x;
  if (i < n) p[i] = rsqrtf(fmaxf(p[i], 1.0f));
}

// ------------------------------------------------------- WMMA fp32 GEMM
// H[N x F] = diag(norm) * A[N x K] @ W[K x F], F = NT*16.
// One wave computes a 16 x F strip; NT v8f accumulators live in VGPRs.
template <int NT>
__global__ __launch_bounds__(256) void gemm_norm_kernel(
    const float* __restrict__ A, const float* __restrict__ W,
    const float* __restrict__ norm, float* __restrict__ H, int N, int K) {
  const int F = NT * 16;
  int gtid = blockIdx.x * blockDim.x + threadIdx.x;
  int wave = gtid >> 5;
  int lane = threadIdx.x & 31;
  int m0 = wave * 16;
  if (m0 >= N) return;  // uniform per wave -> EXEC all-ones inside

  int half = lane >> 4;   // 0: K pair {0,1}, 1: K pair {2,3}
  int ml   = lane & 15;
  int row  = m0 + ml;
  float ns = norm[row];
  const float* __restrict__ arow = A + (size_t)row * K;

  v8f acc[NT];
  const v8f vzero = {0.f, 0.f, 0.f, 0.f, 0.f, 0.f, 0.f, 0.f};
#pragma unroll
  for (int t = 0; t < NT; ++t) acc[t] = vzero;

  for (int k0 = 0; k0 < K; k0 += 4) {
    int ka = k0 + half * 2;
    v2f a;
    a.x = arow[ka] * ns;
    a.y = arow[ka + 1] * ns;
    const float* __restrict__ wr = W + (size_t)ka * F;  // rows ka, ka+1
#pragma unroll
    for (int t = 0; t < NT; ++t) {
      int col = t * 16 + ml;
      v2f b;
      b.x = wr[col];
      b.y = wr[F + col];
      acc[t] = __builtin_amdgcn_wmma_f32_16x16x4_f32(
          false, a, false, b, (short)0, acc[t], false, false);
    }
  }

#pragma unroll
  for (int t = 0; t < NT; ++t) {
#pragma unroll
    for (int v = 0; v < 8; ++v) {
      int r = m0 + v + half * 8;   // C/D layout: VGPR v -> M=v | M=v+8
      int c = t * 16 + ml;
      H[(size_t)r * F + c] = acc[t][v];
    }
  }
}

// ----------------------------------------------- edge gather / scatter-add
// Each thread handles one float4 chunk of one edge. cpe = F/4 = 1<<shift.
__global__ void scatter_kernel(const float* __restrict__ H, const int* __restrict__ src,
                               const int* __restrict__ dst, float* __restrict__ agg,
                               int E, int F, int shift) {
  int tid = blockIdx.x * blockDim.x + threadIdx.x;
  int e = tid >> shift;
  if (e >= E) return;
  int c = tid & ((1 << shift) - 1);
  const float4 v = *(const float4*)(H + (size_t)src[e] * F + c * 4);
  float* o = agg + (size_t)dst[e] * F + c * 4;
  atomicAdd(o + 0, v.x);
  atomicAdd(o + 1, v.y);
  atomicAdd(o + 2, v.z);
  atomicAdd(o + 3, v.w);
}

// ------------------------------------------------ bias + dst-norm + relu
__global__ void finalize_kernel(const float* __restrict__ agg, const float* __restrict__ ndst,
                                const float* __restrict__ bias, float* __restrict__ out,
                                int N, int F, int shift, int relu) {
  int tid = blockIdx.x * blockDim.x + threadIdx.x;
  if (tid >= N * F) return;
  int n = tid >> shift;
  int f = tid & (F - 1);
  float v = agg[tid] * ndst[n] + bias[f];
  out[tid] = relu ? fmaxf(v, 0.0f) : v;
}

// ----------------------------------------- device-side embedding select
__global__ void copy_hidden_kernel(const float* __restrict__ h1, const float* __restrict__ h2,
                                   const float* __restrict__ h3, const int* __restrict__ emb,
                                   float* __restrict__ out, int n) {
  int i = blockIdx.x * blockDim.x + threadIdx.x;
  if (i >= n) return;
  int el = emb[0];
  const float* s = (el == 1) ? h1 : (el == 2) ? h2 : h3;
  out[i] = s[i];
}

// --------------------------------------------------------------- launcher
extern "C" void kernel_launch(void* const* d_in, const int* in_sizes, int n_in,
                              void* d_out, int out_size, void* d_ws, size_t ws_size,
                              hipStream_t stream) {
  const float* x  = (const float*)d_in[0];
  const int*   src = (const int*)d_in[1];
  const int*   dst = (const int*)d_in[2];
  const float* W1 = (const float*)d_in[3];
  const float* b1 = (const float*)d_in[4];
  const float* W2 = (const float*)d_in[5];
  const float* b2 = (const float*)d_in[6];
  const float* W3 = (const float*)d_in[7];
  const float* b3 = (const float*)d_in[8];
  const float* W4 = (const float*)d_in[9];
  const float* b4 = (const float*)d_in[10];
  const int*   emb = (const int*)d_in[11];

  float* out = (float*)d_out;
  float* h4_out  = out;                                   // [N, 64]
  float* hid_out = out + (size_t)N_NODES * OUT_F;         // [N, 128]
  float* x_out   = hid_out + (size_t)N_NODES * H_F;       // [N, 512]

  float* ws   = (float*)d_ws;
  float* nsrc = ws;                 // NPAD
  float* ndst = nsrc + NPAD;        // NPAD
  float* h1   = ndst + NPAD;        // N*H_F
  float* h2   = h1 + (size_t)N_NODES * H_F;
  float* h3   = h2 + (size_t)N_NODES * H_F;
  float* Ht   = h3 + (size_t)N_NODES * H_F;  // pre-agg buffer (max width 128)
  float* agg  = Ht + (size_t)N_NODES * H_F;  // aggregation buffer

  const int B = 256;
  auto nb = [](long long n, int b) { return (int)((n + b - 1) / b); };

  // ---- degrees & norms (deg accumulated in-place, then rsqrt(max(.,1)))
  zero_kernel<<<nb(2 * NPAD, B), B, 0, stream>>>(nsrc, 2 * NPAD);
  degree_kernel<<<nb(N_EDGES, B), B, 0, stream>>>(src, dst, nsrc, ndst, N_EDGES);
  norm_kernel<<<nb(2 * NPAD, B), B, 0, stream>>>(nsrc, 2 * NPAD);

  const int gemm_threads = (N_NODES / 16) * 32;  // one wave per 16-row strip
  const int NH = N_NODES * H_F;
  const int NO = N_NODES * OUT_F;

  // ---- Layer 1: 512 -> 128, relu
  gemm_norm_kernel<8><<<nb(gemm_threads, B), B, 0, stream>>>(x, W1, nsrc, Ht, N_NODES, IN_F);
  zero_kernel<<<nb(NH, B), B, 0, stream>>>(agg, NH);
  scatter_kernel<<<nb((long long)N_EDGES * 32, B), B, 0, stream>>>(Ht, src, dst, agg, N_EDGES, H_F, 5);
  finalize_kernel<<<nb(NH, B), B, 0, stream>>>(agg, ndst, b1, h1, N_NODES, H_F, 7, 1);

  // ---- Layer 2: 128 -> 128, relu
  gemm_norm_kernel<8><<<nb(gemm_threads, B), B, 0, stream>>>(h1, W2, nsrc, Ht, N_NODES, H_F);
  zero_kernel<<<nb(NH, B), B, 0, stream>>>(agg, NH);
  scatter_kernel<<<nb((long long)N_EDGES * 32, B), B, 0, stream>>>(Ht, src, dst, agg, N_EDGES, H_F, 5);
  finalize_kernel<<<nb(NH, B), B, 0, stream>>>(agg, ndst, b2, h2, N_NODES, H_F, 7, 1);

  // ---- Layer 3: 128 -> 128, relu
  gemm_norm_kernel<8><<<nb(gemm_threads, B), B, 0, stream>>>(h2, W3, nsrc, Ht, N_NODES, H_F);
  zero_kernel<<<nb(NH, B), B, 0, stream>>>(agg, NH);
  scatter_kernel<<<nb((long long)N_EDGES * 32, B), B, 0, stream>>>(Ht, src, dst, agg, N_EDGES, H_F, 5);
  finalize_kernel<<<nb(NH, B), B, 0, stream>>>(agg, ndst, b3, h3, N_NODES, H_F, 7, 1);

  // ---- Layer 4: 128 -> 64, no relu, straight to d_out
  gemm_norm_kernel<4><<<nb(gemm_threads, B), B, 0, stream>>>(h3, W4, nsrc, Ht, N_NODES, H_F);
  zero_kernel<<<nb(NO, B), B, 0, stream>>>(agg, NO);
  scatter_kernel<<<nb((long long)N_EDGES * 16, B), B, 0, stream>>>(Ht, src, dst, agg, N_EDGES, OUT_F, 4);
  finalize_kernel<<<nb(NO, B), B, 0, stream>>>(agg, ndst, b4, h4_out, N_NODES, OUT_F, 6, 0);

  // ---- hidden embedding (device-side select, capture-safe) + x passthrough
  copy_hidden_kernel<<<nb(NH, B), B, 0, stream>>>(h1, h2, h3, emb, hid_out, NH);
  hipMemcpyAsync(x_out, x, (size_t)N_NODES * IN_F * sizeof(float),
                 hipMemcpyDeviceToDevice, stream);
}